// RangeStep_67808943669345
// MI455X (gfx1250) — compile-verified
//
#include <hip/hip_runtime.h>
#include <hip/hip_bf16.h>

// N=8, C=64, H=W=128, HW=16384, per-n patch-flat length = 64*16384*9 = 9437184
// T[n,m] = sum of 64 consecutive patch-flat elems at 64m ; m in [0,147456)
// M[n,p] = sum_c x[n,c,p]
// out[n,c,h,w] = softsum[n, c*256 + 2h + (w>>6)] + boxsum3x3(x)[n,c,h,w]

typedef __attribute__((ext_vector_type(2))) float v2f;
typedef __attribute__((ext_vector_type(8))) float v8f;

// ---------------- Kernel 1: M[n,p] via f32 WMMA (ones B-matrix) -------------
__global__ __launch_bounds__(256) void kM(const float* __restrict__ x,
                                          float* __restrict__ M) {
    const int lane = threadIdx.x & 31;
    const int wave = (blockIdx.x * 256 + threadIdx.x) >> 5;  // 0..8191
    const int n  = wave >> 10;            // 1024 tiles of 16 positions per n
    const int p0 = (wave & 1023) << 4;    // base position of this 16-wide tile
    const int i    = lane & 15;           // A-matrix row (M index)
    const int half = lane >> 4;           // 0: K=0,1  1: K=2,3  (ISA A layout)

    const float* xb = x + ((size_t)n * 64) * 16384 + p0 + i;

    v8f c = {};                           // C/D accumulator (16x16 f32)
    v2f b; b.x = 1.0f; b.y = 1.0f;        // B = ones(4x16): D[i][j] = sum_k A[i][k]
    for (int kb = 0; kb < 64; kb += 4) {
        v2f a;
        a.x = xb[(size_t)(kb + 2 * half)     * 16384];
        a.y = xb[(size_t)(kb + 2 * half + 1) * 16384];
        c = __builtin_amdgcn_wmma_f32_16x16x4_f32(
            /*neg_a=*/false, a, /*neg_b=*/false, b,
            /*c_mod=*/(short)0, c, /*reuse_a=*/false, /*reuse_b=*/false);
    }
    // D layout: VGPR r, lanes 0-15 -> D[r][lane]; lanes 16-31 -> D[r+8][lane-16]
    // Column 0 lives in lane 0 (rows 0..7) and lane 16 (rows 8..15).
    float* Mb = M + (size_t)n * 16384 + p0;
    if (lane == 0) {
        for (int r = 0; r < 8; ++r) Mb[r] = c[r];
    } else if (lane == 16) {
        for (int r = 0; r < 8; ++r) Mb[8 + r] = c[r];
    }
}

// ---------------- Kernel 2: T block-64 sums, LDS-staged ---------------------
// grid: n*64*9 blocks; block handles 256 consecutive blocks (m') in one channel
__global__ __launch_bounds__(256) void kT(const float* __restrict__ x,
                                          float* __restrict__ T) {
    const int bid   = blockIdx.x;
    const int chunk = bid % 9;
    const int ci    = (bid / 9) % 64;
    const int n     = bid / (9 * 64);

    const int t0base   = 16384 * chunk;            // first flat elem of chunk
    const int pi_start = t0base / 9;
    const int pi_end   = (t0base + 16383) / 9;
    const int r_lo  = pi_start >> 7;
    const int r_hi  = pi_end >> 7;
    const int nrows = r_hi - r_lo + 3;             // + halo rows (<= 18)

    __shared__ float tile[18][130];                // rows r_lo-1.., cols -1..128
    const float* xb = x + ((size_t)n * 64 + ci) * 16384;
    for (int idx = threadIdx.x; idx < nrows * 130; idx += 256) {
        int lr  = idx / 130;
        int lc  = idx - lr * 130;
        int r   = r_lo - 1 + lr;
        int col = lc - 1;
        float v = 0.0f;
        if ((unsigned)r < 128u && (unsigned)col < 128u) v = xb[r * 128 + col];
        tile[lr][lc] = v;
    }
    __syncthreads();

    const int mprime = chunk * 256 + threadIdx.x;  // block index in channel
    const int t0 = 64 * mprime;
    float s = 0.0f;
    for (int j = 0; j < 64; ++j) {
        int tj   = t0 + j;
        int pi   = tj / 9;              // position
        int srem = tj - pi * 9;         // window element 0..8
        int a    = srem / 3;
        int bb   = srem - a * 3;
        s += tile[(pi >> 7) + a - r_lo][(pi & 127) + bb];
    }
    T[(size_t)n * 147456 + ci * 2304 + mprime] = s;
}

// ---------------- Kernel 3: per-(n,s) softmax normalizers -------------------
__global__ __launch_bounds__(256) void kRed(const float* __restrict__ T,
                                            const float* __restrict__ M,
                                            float* __restrict__ mx,
                                            float* __restrict__ se) {
    const int n = blockIdx.x / 9;
    const int s = blockIdx.x % 9;
    const float* Tb = T + (size_t)n * 147456 + s;
    const float* Mb = M + (size_t)n * 16384;
    __shared__ float red[256];

    float lmax = -3.402823466e38f;
    for (int p = threadIdx.x; p < 16384; p += 256) {
        float v = Tb[9 * p] + Mb[p];
        lmax = fmaxf(lmax, v);
    }
    red[threadIdx.x] = lmax;
    __syncthreads();
    for (int off = 128; off > 0; off >>= 1) {
        if ((int)threadIdx.x < off)
            red[threadIdx.x] = fmaxf(red[threadIdx.x], red[threadIdx.x + off]);
        __syncthreads();
    }
    const float m = red[0];
    __syncthreads();

    float lsum = 0.0f;
    for (int p = threadIdx.x; p < 16384; p += 256) {
        float v = Tb[9 * p] + Mb[p];
        lsum += __expf(v - m);
    }
    red[threadIdx.x] = lsum;
    __syncthreads();
    for (int off = 128; off > 0; off >>= 1) {
        if ((int)threadIdx.x < off)
            red[threadIdx.x] = red[threadIdx.x] + red[threadIdx.x + off];
        __syncthreads();
    }
    if (threadIdx.x == 0) {
        mx[blockIdx.x] = m;
        se[blockIdx.x] = red[0];
    }
}

// ---------------- Kernel 4: softsum[n,p] = sum_s exp(v_s - mx_s)/se_s -------
__global__ __launch_bounds__(256) void kSoftsum(const float* __restrict__ T,
                                                const float* __restrict__ M,
                                                const float* __restrict__ mx,
                                                const float* __restrict__ se,
                                                float* __restrict__ softsum) {
    const int idx = blockIdx.x * 256 + threadIdx.x;  // 0..131071
    const int n = idx >> 14;                         // block fully inside one n
    const int p = idx & 16383;
    __shared__ float smx[9], sinv[9];
    if (threadIdx.x < 9) {
        smx[threadIdx.x]  = mx[n * 9 + threadIdx.x];
        sinv[threadIdx.x] = 1.0f / se[n * 9 + threadIdx.x];
    }
    __syncthreads();
    const float* Tb = T + (size_t)n * 147456 + 9 * (size_t)p;
    const float m = M[idx];
    float acc = 0.0f;
    for (int s = 0; s < 9; ++s)
        acc += __expf(Tb[s] + m - smx[s]) * sinv[s];
    softsum[idx] = acc;
}

// ---------------- Kernel 5: out = boxsum3x3(x) + broadcast(softsum) ---------
// grid: n*64*16 blocks; block = one channel image, 8 rows
__global__ __launch_bounds__(256) void kOut(const float* __restrict__ x,
                                            const float* __restrict__ softsum,
                                            float* __restrict__ out) {
    const int bid = blockIdx.x;
    const int rc = bid & 15;
    const int c  = (bid >> 4) & 63;
    const int n  = bid >> 10;
    const int r0 = rc * 8;

    __shared__ float tile[10][130];
    const float* xb = x + ((size_t)n * 64 + c) * 16384;
    for (int idx = threadIdx.x; idx < 10 * 130; idx += 256) {
        int lr  = idx / 130;
        int lc  = idx - lr * 130;
        int r   = r0 - 1 + lr;
        int col = lc - 1;
        float v = 0.0f;
        if ((unsigned)r < 128u && (unsigned)col < 128u) v = xb[r * 128 + col];
        tile[lr][lc] = v;
    }
    __syncthreads();

    const float* ssb = softsum + (size_t)n * 16384;
    float* ob = out + ((size_t)n * 64 + c) * 16384;
    for (int k = threadIdx.x; k < 1024; k += 256) {
        int lr = k >> 7;            // row within chunk
        int wv = k & 127;
        int h  = r0 + lr;
        float b = tile[lr    ][wv] + tile[lr    ][wv + 1] + tile[lr    ][wv + 2]
                + tile[lr + 1][wv] + tile[lr + 1][wv + 1] + tile[lr + 1][wv + 2]
                + tile[lr + 2][wv] + tile[lr + 2][wv + 1] + tile[lr + 2][wv + 2];
        int pout = c * 256 + h * 2 + (wv >> 6);
        ob[h * 128 + wv] = ssb[pout] + b;
    }
}

extern "C" void kernel_launch(void* const* d_in, const int* in_sizes, int n_in,
                              void* d_out, int out_size, void* d_ws, size_t ws_size,
                              hipStream_t stream) {
    const float* x = (const float*)d_in[0];
    float* out = (float*)d_out;
    float* ws  = (float*)d_ws;

    float* T       = ws;                               // 8*147456 = 1,179,648 f32
    float* M       = T + (size_t)8 * 147456;           // 131,072 f32
    float* mx      = M + (size_t)8 * 16384;            // 72 f32
    float* se      = mx + 72;                          // 72 f32
    float* softsum = se + 72;                          // 131,072 f32  (~5.8 MB total)

    kM      <<<1024,       256, 0, stream>>>(x, M);
    kT      <<<8 * 64 * 9, 256, 0, stream>>>(x, T);
    kRed    <<<72,         256, 0, stream>>>(T, M, mx, se);
    kSoftsum<<<512,        256, 0, stream>>>(T, M, mx, se, softsum);
    kOut    <<<8192,       256, 0, stream>>>(x, softsum, out);
}